// Ca_Aware_Embedder_76450417869193
// MI455X (gfx1250) — compile-verified
//
#include <hip/hip_runtime.h>

typedef __attribute__((ext_vector_type(2))) float v2f;
typedef __attribute__((ext_vector_type(8))) float v8f;

#define NRES   1024
#define CZ     128
#define NBINS  15
#define KPAD   16          // K padded to 16 (zero row at k=15)
#define INF_F  100000000.0f

// Each wave computes a 16x128 strip of the [NRES*NRES x CZ] output:
//   rows  = 16 consecutive flattened (i,j) pairs
//   cols  = all 128 channels (8 column tiles of 16)
// via V_WMMA_F32_16X16X4_F32 chained over 4 K-chunks (K=16, padded).
// Output is a 512MB pure write stream (never re-read) -> non-temporal stores.
__global__ __launch_bounds__(256) void ca_embed_wmma_kernel(
    const float* __restrict__ x,     // [NRES,3]
    const float* __restrict__ W,     // [CZ, NBINS] (torch Linear weight [out,in])
    const float* __restrict__ bvec,  // [CZ]
    float* __restrict__ out)         // [NRES*NRES, CZ]
{
    // ---- Stage W into LDS, padded to KPAD with a zero row at k=15 ----
    __shared__ float ldsW[CZ * KPAD];         // [c][k], 8 KB
    const int tid = threadIdx.x;
    for (int idx = tid; idx < CZ * KPAD; idx += 256) {
        const int c = idx >> 4;
        const int k = idx & (KPAD - 1);
        ldsW[idx] = (k < NBINS) ? W[c * NBINS + k] : 0.0f;
    }
    __syncthreads();

    const int wave = tid >> 5;                 // 0..7
    const int lane = tid & 31;
    const unsigned tile_m = blockIdx.x * 8u + wave;   // 0..65535
    const unsigned P0 = tile_m * 16u;          // base flattened pair index

    const int m     = lane & 15;               // row within tile this lane describes for A
    const int khalf = (lane >> 4) << 1;        // 0 for lanes 0-15, 2 for lanes 16-31
    const int n     = lane & 15;               // column within tile for B/D

    // ---- Per-row squared distance (two lanes redundantly compute same row) ----
    const unsigned p = P0 + (unsigned)m;
    const int i = (int)(p >> 10);
    const int j = (int)(p & 1023u);

    const float dx = x[i * 3 + 0] - x[j * 3 + 0];
    const float dy = x[i * 3 + 1] - x[j * 3 + 1];
    const float dz = x[i * 3 + 2] - x[j * 3 + 2];
    const float d  = dx * dx + dy * dy + dz * dz;

    // ---- Bin index with exactly the reference's strict-inequality semantics ----
    int bidx = -1;
    #pragma unroll
    for (int k = 0; k < NBINS; ++k) {
        const float blo = 3.25f + 1.25f * (float)k;
        const float bhi = 3.25f + 1.25f * (float)(k + 1);
        const float lo  = blo * blo;
        const float hi  = (k == NBINS - 1) ? INF_F : bhi * bhi;
        if (d > lo && d < hi) bidx = k;
    }

    // ---- Build fp32 one-hot A tiles: 16x4 per chunk, 4 chunks (K = 0..15) ----
    // Layout (32-bit A 16x4): lanes 0-15 hold K=c*4+{0,1}; lanes 16-31 K=c*4+{2,3}.
    v2f A[4];
    #pragma unroll
    for (int c = 0; c < 4; ++c) {
        const int k0 = 4 * c + khalf;
        A[c].x = (bidx == k0)     ? 1.0f : 0.0f;
        A[c].y = (bidx == k0 + 1) ? 1.0f : 0.0f;
    }

    // ---- 8 column tiles of 16 channels each ----
    #pragma unroll
    for (int t = 0; t < 8; ++t) {
        const int c0 = t * 16;
        const float bval = bvec[c0 + n];

        v8f acc;
        #pragma unroll
        for (int v = 0; v < 8; ++v) acc[v] = bval;   // fold bias into C

        #pragma unroll
        for (int c = 0; c < 4; ++c) {
            const int k0 = 4 * c + khalf;            // even -> 8B-aligned LDS pair
            v2f B;
            B.x = ldsW[(c0 + n) * KPAD + k0];
            B.y = ldsW[(c0 + n) * KPAD + k0 + 1];
            acc = __builtin_amdgcn_wmma_f32_16x16x4_f32(
                /*neg_a=*/false, A[c], /*neg_b=*/false, B,
                /*c_mod=*/(short)0, acc, /*reuse_a=*/false, /*reuse_b=*/false);
        }

        // ---- Store D non-temporally: lanes 0-15 -> rows 0-7, lanes 16-31 -> rows 8-15.
        // Per instruction: two contiguous 64B row segments across the wave.
        const unsigned mbase = (lane >> 4) * 8u;
        #pragma unroll
        for (int v = 0; v < 8; ++v) {
            const size_t row = (size_t)P0 + mbase + (unsigned)v;
            __builtin_nontemporal_store(acc[v], &out[row * CZ + (unsigned)(c0 + n)]);
        }
    }
}

extern "C" void kernel_launch(void* const* d_in, const int* in_sizes, int n_in,
                              void* d_out, int out_size, void* d_ws, size_t ws_size,
                              hipStream_t stream) {
    (void)in_sizes; (void)n_in; (void)d_ws; (void)ws_size; (void)out_size;
    const float* x = (const float*)d_in[0];   // [1,1024,3]
    const float* W = (const float*)d_in[1];   // [128,15]
    const float* b = (const float*)d_in[2];   // [128]
    float* out = (float*)d_out;               // [1,1024,1024,128]

    const int total_row_tiles = (NRES * NRES) / 16;   // 65536
    const int blocks = total_row_tiles / 8;           // 8 waves per block
    ca_embed_wmma_kernel<<<blocks, 256, 0, stream>>>(x, W, b, out);
}